// MACELayer_78683800863181
// MI455X (gfx1250) — compile-verified
//
#include <hip/hip_runtime.h>
#include <math.h>
#include <stdint.h>

#define NNODE 8000
#define NEDGE 128000
#define FDIM  128
#define RDIM  8

typedef float v2f __attribute__((ext_vector_type(2)));
typedef float v8f __attribute__((ext_vector_type(8)));

__device__ __forceinline__ v8f wmma_f32_k4(v2f a, v2f b, v8f c) {
    // D = A(16x4) * B(4x16) + C(16x16), fp32, wave32
    return __builtin_amdgcn_wmma_f32_16x16x4_f32(false, a, false, b, (short)0, c, false, false);
}

__device__ __forceinline__ float silu_f(float x) {
    return x / (1.0f + __expf(-x));
}

// ---- CDNA5 async global->LDS copy (ASYNCcnt path), 16B per lane ------------
typedef __attribute__((address_space(3))) char lds_char_t;

__device__ __forceinline__ void async_ld_b128(void* lds_ptr, const void* gptr) {
    uint32_t loff = (uint32_t)(uintptr_t)(lds_char_t*)lds_ptr;   // LDS byte offset
    asm volatile("global_load_async_to_lds_b128 %0, %1, off"
                 :
                 : "v"(loff), "v"(gptr)
                 : "memory");
}
__device__ __forceinline__ void async_wait0() {
    asm volatile("s_wait_asynccnt 0" ::: "memory");
}

// ---------------------------------------------------------------------------
// Generic node-wise linear:  out[n,g] (+)= scale * sum_f in[n,f] * w[f,g]
// STRIDE = 1 (scalar irrep) or 3 (vector irrep; blockIdx.y = component).
// block = 256 threads = 8 waves; wave w owns output columns [16w, 16w+16);
// each block processes 64 rows (4 M-tiles per wave) to amortize W staging.
// W tile staged via GLOBAL_LOAD_ASYNC_TO_LDS_B128.
// ---------------------------------------------------------------------------
template<int STRIDE, bool ACC>
__global__ __launch_bounds__(256)
void node_linear_kernel(const float* __restrict__ in, const float* __restrict__ w,
                        float* __restrict__ out, float scale)
{
    __shared__ float Ash[64][132];          // 64 rows x 128 K, padded (132%64=4 -> bank spread)
    __shared__ float Wsh[FDIM * FDIM];      // full 128x128 weight tile

    const int tid = threadIdx.x;
    if (STRIDE == 3) {                      // vector component handled by this block
        in  += blockIdx.y;
        out += blockIdx.y;
    }

    const int row0 = blockIdx.x * 64;

    // async-stage W: 16384 floats = 4096 x 16B chunks, 16 per thread
    #pragma unroll
    for (int i = 0; i < (FDIM * FDIM) / (4 * 256); ++i) {
        int c16 = tid + i * 256;
        async_ld_b128(&Wsh[c16 * 4], w + c16 * 4);
    }
    // stage A (strided gather; overlapped with the async W copy)
    #pragma unroll
    for (int i = 0; i < (64 * FDIM) / 256; ++i) {
        int idx = tid + i * 256;
        int r = idx >> 7, k = idx & 127;
        Ash[r][k] = in[((size_t)(row0 + r) * FDIM + k) * STRIDE];
    }
    async_wait0();
    __syncthreads();

    const int lane = tid & 31;
    const int wave = tid >> 5;
    const int m    = lane & 15;
    const int half = lane >> 4;
    const int ksel = half * 2;             // lanes 0-15: K=k0,k0+1 ; lanes 16-31: K=k0+2,k0+3
    const int g    = wave * 16 + m;        // B/C column for this lane

    #pragma unroll 1
    for (int mt = 0; mt < 4; ++mt) {
        v8f c = {0, 0, 0, 0, 0, 0, 0, 0};
        #pragma unroll 8
        for (int k0 = 0; k0 < FDIM; k0 += 4) {
            v2f a, b;
            a.x = Ash[mt * 16 + m][k0 + ksel];
            a.y = Ash[mt * 16 + m][k0 + ksel + 1];
            b.x = Wsh[(k0 + ksel) * FDIM + g];
            b.y = Wsh[(k0 + ksel + 1) * FDIM + g];
            c = wmma_f32_k4(a, b, c);
        }
        #pragma unroll
        for (int r = 0; r < 8; ++r) {
            int row = mt * 16 + r + 8 * half;  // C layout: VGPR r = rows r / r+8
            size_t o = ((size_t)(row0 + row) * FDIM + g) * STRIDE;
            float val = c[r] * scale;
            if constexpr (ACC) val += out[o];
            out[o] = val;
        }
    }
}

// ---------------------------------------------------------------------------
// Fused edge kernel: radial MLP (WMMA GEMMs through LDS) + CG tensor product
// + scatter-add.  One wave per 16-edge tile, 8 waves / block.
// w2/w3 staged block-wide via async-to-LDS; sender rows prefetched.
// ---------------------------------------------------------------------------
struct WaveLds {
    float radA[16][9];       // radial [16 x 8], padded row
    float h[2][16][68];      // ping-pong hidden activations, padded rows
    float vhat[16][3];
    int   snd[16];
    int   rcv[16];
};

__global__ __launch_bounds__(256)
void edge_kernel(const float* __restrict__ vectors,
                 const float* __restrict__ radial,
                 const int*   __restrict__ senders,
                 const int*   __restrict__ receivers,
                 const float* __restrict__ x0,
                 const float* __restrict__ x1,
                 const float* __restrict__ w1,   // [8,64]
                 const float* __restrict__ w2,   // [64,64]
                 const float* __restrict__ w3,   // [64,64]
                 const float* __restrict__ w4,   // [64,640]
                 float* __restrict__ a0,
                 float* __restrict__ a1)
{
    __shared__ WaveLds lds[8];
    __shared__ float  W23[2][64 * 64];      // 32KB, shared by all 8 waves

    const int tid  = threadIdx.x;
    const int lane = tid & 31;
    const int wave = tid >> 5;
    WaveLds& L = lds[wave];

    // async-stage w2 and w3 block-wide: 2*4096 floats = 2048 x 16B chunks
    #pragma unroll
    for (int i = 0; i < 8; ++i) {
        int c16 = tid + i * 256;
        if (c16 < 1024) async_ld_b128(&W23[0][c16 * 4], w2 + c16 * 4);
        else            async_ld_b128(&W23[1][(c16 - 1024) * 4], w3 + (c16 - 1024) * 4);
    }

    const int e0 = (blockIdx.x * 8 + wave) * 16;

    if (lane < 16) {
        int e = e0 + lane;
        float vx = vectors[e * 3 + 0], vy = vectors[e * 3 + 1], vz = vectors[e * 3 + 2];
        float nrm = sqrtf(vx * vx + vy * vy + vz * vz);
        float inv = 1.0f / (nrm + 1e-12f);
        L.vhat[lane][0] = vx * inv;
        L.vhat[lane][1] = vy * inv;
        L.vhat[lane][2] = vz * inv;
        const int sn = senders[e];
        L.snd[lane] = sn;
        L.rcv[lane] = receivers[e];
        // warm L2 for the gathered sender rows used by the tensor product
        __builtin_prefetch(x0 + (size_t)sn * FDIM, 0, 0);
        __builtin_prefetch(x1 + (size_t)sn * FDIM * 3, 0, 0);
        __builtin_prefetch(x1 + (size_t)sn * FDIM * 3 + 192, 0, 0);
        #pragma unroll
        for (int k = 0; k < RDIM; ++k)
            L.radA[lane][k] = radial[e * RDIM + k];
    }
    // LDS ops are in-order within a wave: producer/consumer inside one wave is safe.

    const int m    = lane & 15;
    const int half = lane >> 4;
    const int ksel = half * 2;

    const float s1 = 0.35355339059327373f;   // 1/sqrt(R)
    const float s8 = 0.125f;                 // 1/8
    const v8f  vzero = {0, 0, 0, 0, 0, 0, 0, 0};

    // ---- layer 1: [16x8] @ w1[8x64], silu -> h[0]  (overlaps async W23 copy)
    #pragma unroll
    for (int nt = 0; nt < 4; ++nt) {
        v8f c = vzero;
        #pragma unroll
        for (int k0 = 0; k0 < 8; k0 += 4) {
            v2f a, b;
            a.x = L.radA[m][k0 + ksel];
            a.y = L.radA[m][k0 + ksel + 1];
            b.x = w1[(k0 + ksel) * 64 + nt * 16 + m];
            b.y = w1[(k0 + ksel + 1) * 64 + nt * 16 + m];
            c = wmma_f32_k4(a, b, c);
        }
        #pragma unroll
        for (int r = 0; r < 8; ++r)
            L.h[0][r + 8 * half][nt * 16 + m] = silu_f(c[r] * s1);
    }

    // publish async-staged w2/w3 to the whole block
    async_wait0();
    __syncthreads();

    // ---- layers 2 & 3: [16x64] @ [64x64], silu; ping-pong h[0] -> h[1] -> h[0]
    #pragma unroll 1
    for (int layer = 0; layer < 2; ++layer) {
        const float* wlds = &W23[layer][0];
        const int src = layer;       // 0 then 1
        const int dst = 1 - layer;   // 1 then 0
        #pragma unroll
        for (int nt = 0; nt < 4; ++nt) {
            v8f c = vzero;
            #pragma unroll
            for (int k0 = 0; k0 < 64; k0 += 4) {
                v2f a, b;
                a.x = L.h[src][m][k0 + ksel];
                a.y = L.h[src][m][k0 + ksel + 1];
                b.x = wlds[(k0 + ksel) * 64 + nt * 16 + m];
                b.y = wlds[(k0 + ksel + 1) * 64 + nt * 16 + m];
                c = wmma_f32_k4(a, b, c);
            }
            #pragma unroll
            for (int r = 0; r < 8; ++r)
                L.h[dst][r + 8 * half][nt * 16 + m] = silu_f(c[r] * s8);
        }
    }

    // ---- layer 4 + tensor product, per 16-feature tile.
    // mix columns for feature tile t are the 5 N-tiles {p*8+t}: keep all 5 path
    // accumulators live in VGPRs; C layout (row = r+8*half, col = lane%16)
    // directly gives mix[edge, path, feat] for the elementwise CG product.
    #pragma unroll 1
    for (int t = 0; t < 8; ++t) {
        v8f cm[5];
        #pragma unroll
        for (int p = 0; p < 5; ++p) cm[p] = vzero;

        #pragma unroll
        for (int k0 = 0; k0 < 64; k0 += 4) {
            v2f a;
            a.x = L.h[0][m][k0 + ksel];
            a.y = L.h[0][m][k0 + ksel + 1];
            #pragma unroll
            for (int p = 0; p < 5; ++p) {
                v2f b;
                b.x = w4[(k0 + ksel)     * 640 + p * 128 + t * 16 + m];
                b.y = w4[(k0 + ksel + 1) * 640 + p * 128 + t * 16 + m];
                cm[p] = wmma_f32_k4(a, b, cm[p]);
            }
        }

        const int feat = t * 16 + m;
        #pragma unroll
        for (int r = 0; r < 8; ++r) {
            const int   le = r + 8 * half;       // local edge this lane holds in VGPR r
            const int   sn = L.snd[le];
            const int   rc = L.rcv[le];
            const float vx = L.vhat[le][0], vy = L.vhat[le][1], vz = L.vhat[le][2];
            const size_t base = (size_t)sn * FDIM + feat;
            const float m0v = x0[base];
            const float m1x = x1[base * 3 + 0];
            const float m1y = x1[base * 3 + 1];
            const float m1z = x1[base * 3 + 2];
            const float mix0 = cm[0][r] * s8, mix1 = cm[1][r] * s8, mix2 = cm[2][r] * s8;
            const float mix3 = cm[3][r] * s8, mix4 = cm[4][r] * s8;
            // with vhat the sqrt(3) factors of Y1 cancel against /sqrt(3)
            const float dot = m1x * vx + m1y * vy + m1z * vz;
            const float o0  = (mix0 * m0v + mix1 * dot) * 0.25f;   // * EPS
            const float crx = m1y * vz - m1z * vy;
            const float cry = m1z * vx - m1x * vz;
            const float crz = m1x * vy - m1y * vx;
            const float o1x = (mix2 * m0v * vx + mix3 * m1x + mix4 * crx) * 0.25f;
            const float o1y = (mix2 * m0v * vy + mix3 * m1y + mix4 * cry) * 0.25f;
            const float o1z = (mix2 * m0v * vz + mix3 * m1z + mix4 * crz) * 0.25f;
            atomicAdd(&a0[(size_t)rc * FDIM + feat], o0);
            const size_t ob = ((size_t)rc * FDIM + feat) * 3;
            atomicAdd(&a1[ob + 0], o1x);
            atomicAdd(&a1[ob + 1], o1y);
            atomicAdd(&a1[ob + 2], o1z);
        }
    }
}

// ---------------------------------------------------------------------------
// Symmetric contraction (elementwise, species-dependent weights)
// ---------------------------------------------------------------------------
__global__ __launch_bounds__(256)
void symcon_kernel(const float* __restrict__ s, const float* __restrict__ v,
                   const int* __restrict__ species, const float* __restrict__ w_sc,
                   float* __restrict__ b0, float* __restrict__ b1)
{
    const int idx = blockIdx.x * 256 + threadIdx.x;   // [0, N*F)
    const int n = idx >> 7;
    const int f = idx & 127;
    const float sv = s[idx];
    const float vx = v[(size_t)idx * 3 + 0];
    const float vy = v[(size_t)idx * 3 + 1];
    const float vz = v[(size_t)idx * 3 + 2];
    const float vv = vx * vx + vy * vy + vz * vz;
    const int z = species[n];
    const float* w = w_sc + (size_t)z * 9 * FDIM + f;
    const float szi = 0.31622776601683794f;           // 1/sqrt(Z)
    const float w0 = w[0] * szi,        w1 = w[FDIM] * szi,     w2 = w[2 * FDIM] * szi;
    const float w3 = w[3 * FDIM] * szi, w4 = w[4 * FDIM] * szi, w5 = w[5 * FDIM] * szi;
    const float w6 = w[6 * FDIM] * szi, w7 = w[7 * FDIM] * szi, w8 = w[8 * FDIM] * szi;
    const float sq = sv * sv;
    b0[idx] = w0 * sv + w1 * sq + w2 * vv + w3 * sq * sv + w4 * sv * vv;
    const float tc = w5 + w6 * sv + w7 * sq + w8 * vv;
    const size_t o = (size_t)idx * 3;
    b1[o + 0] = tc * vx;
    b1[o + 1] = tc * vy;
    b1[o + 2] = tc * vz;
}

// ---------------------------------------------------------------------------
// linZ self-connection: species-dependent F x F; writes sc directly into the
// f0/f1 output regions (post-linear GEMM accumulates on top).
// ---------------------------------------------------------------------------
__global__ __launch_bounds__(128)
void skipz_kernel(const float* __restrict__ nf0, const float* __restrict__ nf1,
                  const int* __restrict__ species,
                  const float* __restrict__ wz0, const float* __restrict__ wz1,
                  float* __restrict__ f0, float* __restrict__ f1)
{
    __shared__ float r0[FDIM];
    __shared__ float r1[FDIM * 3];
    const int n = blockIdx.x;
    const int g = threadIdx.x;

    r0[g] = nf0[(size_t)n * FDIM + g];
    #pragma unroll
    for (int i = 0; i < 3; ++i)
        r1[g + i * FDIM] = nf1[(size_t)n * FDIM * 3 + g + i * FDIM];
    __syncthreads();

    const int z = species[n];
    const float* w0 = wz0 + (size_t)z * FDIM * FDIM;
    const float* w1 = wz1 + (size_t)z * FDIM * FDIM;
    float acc0 = 0.f, ax = 0.f, ay = 0.f, az = 0.f;
    #pragma unroll 4
    for (int f = 0; f < FDIM; ++f) {
        const float a = w0[f * FDIM + g];
        const float b = w1[f * FDIM + g];
        acc0 += r0[f] * a;
        ax   += r1[f * 3 + 0] * b;
        ay   += r1[f * 3 + 1] * b;
        az   += r1[f * 3 + 2] * b;
    }
    const float sc = 0.027950849718747374f;   // 1/(sqrt(F)*sqrt(Z))
    f0[(size_t)n * FDIM + g] = acc0 * sc;
    const size_t o = ((size_t)n * FDIM + g) * 3;
    f1[o + 0] = ax * sc;
    f1[o + 1] = ay * sc;
    f1[o + 2] = az * sc;
}

// ---------------------------------------------------------------------------
// Readout: out[n] = (f0[n,:] . w_read0) / sqrt(F)
// ---------------------------------------------------------------------------
__global__ __launch_bounds__(128)
void readout_kernel(const float* __restrict__ f0, const float* __restrict__ w_read0,
                    float* __restrict__ out)
{
    __shared__ float red[FDIM];
    const int n = blockIdx.x;
    const int tid = threadIdx.x;
    red[tid] = f0[(size_t)n * FDIM + tid] * w_read0[tid];
    __syncthreads();
    #pragma unroll
    for (int sft = 64; sft > 0; sft >>= 1) {
        if (tid < sft) red[tid] += red[tid + sft];
        __syncthreads();
    }
    if (tid == 0) out[n] = red[0] * 0.08838834764831845f;   // 1/sqrt(F)
}

// ---------------------------------------------------------------------------
extern "C" void kernel_launch(void* const* d_in, const int* in_sizes, int n_in,
                              void* d_out, int out_size, void* d_ws, size_t ws_size,
                              hipStream_t stream)
{
    const float* vectors   = (const float*)d_in[0];
    const float* nf0       = (const float*)d_in[1];
    const float* nf1       = (const float*)d_in[2];
    const float* radial    = (const float*)d_in[3];
    const int*   species   = (const int*)d_in[4];
    const int*   senders   = (const int*)d_in[5];
    const int*   receivers = (const int*)d_in[6];
    const float* w_skipZ0  = (const float*)d_in[7];
    const float* w_skipZ1  = (const float*)d_in[8];
    const float* w_up0     = (const float*)d_in[9];
    const float* w_up1     = (const float*)d_in[10];
    const float* mlp_w1    = (const float*)d_in[11];
    const float* mlp_w2    = (const float*)d_in[12];
    const float* mlp_w3    = (const float*)d_in[13];
    const float* mlp_w4    = (const float*)d_in[14];
    const float* w_down0   = (const float*)d_in[15];
    const float* w_down1   = (const float*)d_in[16];
    const float* w_sc      = (const float*)d_in[17];
    const float* w_post0   = (const float*)d_in[18];
    const float* w_post1   = (const float*)d_in[19];
    const float* w_read0   = (const float*)d_in[20];

    const size_t NF = (size_t)NNODE * FDIM;
    float* ws = (float*)d_ws;
    float* x0 = ws;             // [N,F]   ; reused as s after edge kernel
    float* x1 = ws + NF;        // [N,F,3] ; reused as v
    float* a0 = ws + 4 * NF;    // [N,F]   ; reused as b0
    float* a1 = ws + 5 * NF;    // [N,F,3] ; reused as b1

    float* out      = (float*)d_out;
    float* out_read = out;                  // [N,1]
    float* f0       = out + NNODE;          // [N,F]
    float* f1       = out + NNODE + NF;     // [N,F,3]

    const float invSqF = 0.08838834764831845f;   // 1/sqrt(F)
    const dim3 gS(NNODE / 64, 1);                // scalar-irrep GEMM grid
    const dim3 gV(NNODE / 64, 3);                // vector-irrep GEMM grid (per component)

    // zero edge-sum accumulators (graph-capturable)
    hipMemsetAsync(a0, 0, 4 * NF * sizeof(float), stream);

    // linear_up: x = feats @ w_up / sqrt(F)
    node_linear_kernel<1, false><<<gS, 256, 0, stream>>>(nf0, w_up0, x0, invSqF);
    node_linear_kernel<3, false><<<gV, 256, 0, stream>>>(nf1, w_up1, x1, invSqF);

    // fused radial-MLP + CG tensor product + scatter
    edge_kernel<<<NEDGE / 128, 256, 0, stream>>>(
        vectors, radial, senders, receivers, x0, x1,
        mlp_w1, mlp_w2, mlp_w3, mlp_w4, a0, a1);

    // linear_down: s,v = a @ w_down / sqrt(F)   (writes into dead x0/x1 region)
    node_linear_kernel<1, false><<<gS, 256, 0, stream>>>(a0, w_down0, x0, invSqF);
    node_linear_kernel<3, false><<<gV, 256, 0, stream>>>(a1, w_down1, x1, invSqF);

    // symmetric contraction -> b0/b1 (overwrites dead a0/a1 region)
    symcon_kernel<<<(NNODE * FDIM) / 256, 256, 0, stream>>>(
        x0, x1, species, w_sc, a0, a1);

    // self-connection written straight into output f0/f1
    skipz_kernel<<<NNODE, 128, 0, stream>>>(
        nf0, nf1, species, w_skipZ0, w_skipZ1, f0, f1);

    // linear_post_sc accumulated on top of the self-connection
    node_linear_kernel<1, true><<<gS, 256, 0, stream>>>(a0, w_post0, f0, invSqF);
    node_linear_kernel<3, true><<<gV, 256, 0, stream>>>(a1, w_post1, f1, invSqF);

    // readout
    readout_kernel<<<NNODE, 128, 0, stream>>>(f0, w_read0, out_read);
}